// DynamicNeuralNetworkNAS_63007170233017
// MI455X (gfx1250) — compile-verified
//
#include <hip/hip_runtime.h>

typedef __attribute__((ext_vector_type(2))) float        v2f;
typedef __attribute__((ext_vector_type(8))) float        v8f;
typedef __attribute__((ext_vector_type(4))) unsigned int v4u;
typedef __attribute__((ext_vector_type(8))) int          v8i;
typedef __attribute__((ext_vector_type(4))) int          v4i;

#define KDIM     64
#define PITCH    66          // LDS pitch in floats (bank-conflict pad)
#define NLAYER   8
#define EPS_LN   1e-5f

__device__ __forceinline__ unsigned lds_off(const void* p) {
  return (unsigned)(unsigned long long)(size_t)p;
}

// TDM: contiguous 4096-float tile (one layer's 64x64 weights) -> LDS,
// re-padded on the fly to pitch-66 rows (pad 2 dwords every 64 dwords).
__device__ __forceinline__ void tdm_load_w(const float* gsrc, unsigned dst_lds) {
  unsigned long long ga = (unsigned long long)(size_t)gsrc;
  v4u g0;
  g0[0] = 1u;                                                 // count=1 (valid)
  g0[1] = dst_lds;                                            // lds_addr (bytes)
  g0[2] = (unsigned)ga;                                       // global_addr[31:0]
  g0[3] = (unsigned)((ga >> 32) & 0x01FFFFFFu) | (2u << 30);  // addr[56:32] | type=2
  v8i g1;
  // data_size=4B(code2) | pad_enable | pad_interval=64dw(code5) | pad_amount=2dw(code1)
  g1[0] = (2 << 16) | (1 << 20) | (5 << 22) | (1 << 25);
  g1[1] = (int)(4096u << 16);   // tensor_dim0 = 4096 (bits 79:48, low half)
  g1[2] = (int)(1u << 16);      // tensor_dim1 = 1
  g1[3] = (int)(4096u << 16);   // tile_dim0   = 4096
  g1[4] = 1;                    // tile_dim1   = 1
  g1[5] = 4096;                 // tensor_dim0_stride
  g1[6] = 0;
  g1[7] = 0;
  v4i gz4 = {0, 0, 0, 0};
  v8i gz8 = {0, 0, 0, 0, 0, 0, 0, 0};
  __builtin_amdgcn_tensor_load_to_lds(g0, g1, gz4, gz4, gz8, 0);
}

__global__ __launch_bounds__(256)
void nas_fused_kernel(const float* __restrict__ weights,
                      const float* __restrict__ x,
                      const float* __restrict__ gamma,
                      const float* __restrict__ beta,
                      float* __restrict__ out) {
  __shared__ float Wbuf[2][KDIM * PITCH];
  __shared__ float Xt[KDIM * PITCH];     // activations TRANSPOSED: Xt[n][k]
  __shared__ float redS[8];
  __shared__ float redQ[8];

  const int tid  = threadIdx.x;
  const int lane = tid & 31;
  const int wid  = tid >> 5;
  const int half = lane >> 4;     // 0: lanes 0-15, 1: lanes 16-31
  const int l16  = lane & 15;
  const int b    = blockIdx.x;

  const int tm  = wid >> 1;             // tile row  (0..3)
  const int tn0 = (wid & 1) * 2;        // tile cols (pair)
  const int n0  = tn0 * 16 + l16;
  const int n1  = n0 + 16;
  const int r0  = tm * 16;

  const float* wsrc = weights + (size_t)b * (NLAYER * KDIM * KDIM);
  const float* xsrc = x + (size_t)b * (KDIM * KDIM);
  float*       odst = out + (size_t)b * (KDIM * KDIM);

  // Kick layer-0 weight DMA immediately (wave 0 owns the TDM).
  if (wid == 0) tdm_load_w(wsrc, lds_off(&Wbuf[0][0]));

  // Stage x[b] (16 KB) into LDS, transposed: Xt[s][k] = x[k][s].
  #pragma unroll
  for (int j = 0; j < 4; ++j) {
    int e = (j * 256 + tid) * 4;            // linear element in row-major x
    int k = e >> 6;                         // row of x
    int s = e & 63;                         // col of x (4 consecutive)
    float4 v = reinterpret_cast<const float4*>(xsrc)[j * 256 + tid];
    Xt[(s + 0) * PITCH + k] = v.x;
    Xt[(s + 1) * PITCH + k] = v.y;
    Xt[(s + 2) * PITCH + k] = v.z;
    Xt[(s + 3) * PITCH + k] = v.w;
  }

  // gamma/beta fragments in registers (C/D layout positions are layer-invariant).
  float ga0[8], ga1[8], be0[8], be1[8];
  #pragma unroll
  for (int g = 0; g < 8; ++g) {
    int row = r0 + g + 8 * half;
    ga0[g] = gamma[row * 64 + n0];
    ga1[g] = gamma[row * 64 + n1];
    be0[g] = beta[row * 64 + n0];
    be1[g] = beta[row * 64 + n1];
  }

  if (wid == 0) __builtin_amdgcn_s_wait_tensorcnt(0);
  __syncthreads();

  for (int layer = 0; layer < NLAYER; ++layer) {
    const int cur = layer & 1;
    // Prefetch next layer's weights into the other buffer (overlaps compute).
    if (wid == 0 && layer + 1 < NLAYER)
      tdm_load_w(wsrc + (size_t)(layer + 1) * 4096, lds_off(&Wbuf[cur ^ 1][0]));

    const float* Ws = &Wbuf[cur][0];
    v8f acc0 = {};
    v8f acc1 = {};
    // A fragment: W[r0+l16][K], K = 4*kk + 2*half + j  (contiguous pair)
    const float* arow = &Ws[(r0 + l16) * PITCH + 2 * half];
    // B fragment: {X[kb][n], X[kb+1][n]} = contiguous pair in transposed Xt
    const float* brow0 = &Xt[n0 * PITCH + 2 * half];
    const float* brow1 = &Xt[n1 * PITCH + 2 * half];
    #pragma unroll
    for (int kk = 0; kk < 16; ++kk) {
      v2f a   = *reinterpret_cast<const v2f*>(&arow[kk * 4]);
      v2f bv0 = *reinterpret_cast<const v2f*>(&brow0[kk * 4]);
      v2f bv1 = *reinterpret_cast<const v2f*>(&brow1[kk * 4]);
      acc0 = __builtin_amdgcn_wmma_f32_16x16x4_f32(false, a, false, bv0,
                                                   (short)0, acc0, false, false);
      acc1 = __builtin_amdgcn_wmma_f32_16x16x4_f32(false, a, false, bv1,
                                                   (short)0, acc1, false, false);
    }

    // ---- layernorm over all 4096 elements (workgroup reduction) ----
    float s = 0.f, q = 0.f;
    #pragma unroll
    for (int g = 0; g < 8; ++g) {
      s += acc0[g] + acc1[g];
      q += acc0[g] * acc0[g] + acc1[g] * acc1[g];
    }
    #pragma unroll
    for (int off = 16; off > 0; off >>= 1) {
      s += __shfl_xor(s, off, 32);
      q += __shfl_xor(q, off, 32);
    }
    if (lane == 0) { redS[wid] = s; redQ[wid] = q; }
    __syncthreads();                       // also fences all GEMM reads of Xt
    float S = 0.f, Q = 0.f;
    #pragma unroll
    for (int wv = 0; wv < 8; ++wv) { S += redS[wv]; Q += redQ[wv]; }
    const float mean = S * (1.f / 4096.f);
    const float var  = Q * (1.f / 4096.f) - mean * mean;
    const float rs   = rsqrtf(var + EPS_LN);

    const bool relu = (layer & 1) == 0;
    #pragma unroll
    for (int g = 0; g < 8; ++g) {
      const int row = r0 + g + 8 * half;
      float v0 = (acc0[g] - mean) * rs * ga0[g] + be0[g];
      float v1 = (acc1[g] - mean) * rs * ga1[g] + be1[g];
      if (relu) { v0 = fmaxf(v0, 0.f); v1 = fmaxf(v1, 0.f); }
      if (layer > 0) {                         // residual (each thread owns its slots)
        v0 += Xt[n0 * PITCH + row];
        v1 += Xt[n1 * PITCH + row];
      }
      if (layer < NLAYER - 1) {
        Xt[n0 * PITCH + row] = v0;
        Xt[n1 * PITCH + row] = v1;
      } else {
        odst[row * 64 + n0] = v0;
        odst[row * 64 + n1] = v1;
      }
    }

    if (layer + 1 < NLAYER) {
      if (wid == 0) __builtin_amdgcn_s_wait_tensorcnt(0);  // next W tile landed
      __syncthreads();
    }
  }
}

extern "C" void kernel_launch(void* const* d_in, const int* in_sizes, int n_in,
                              void* d_out, int out_size, void* d_ws, size_t ws_size,
                              hipStream_t stream) {
  const float* weights = (const float*)d_in[0];
  const float* x       = (const float*)d_in[1];
  const float* gamma   = (const float*)d_in[2];
  const float* beta    = (const float*)d_in[3];
  float*       out     = (float*)d_out;

  const int batch = in_sizes[1] / (KDIM * KDIM);   // 2048
  nas_fused_kernel<<<batch, 256, 0, stream>>>(weights, x, gamma, beta, out);
}